// ChamferDistanceLoss_29437705847564
// MI455X (gfx1250) — compile-verified
//
#include <hip/hip_runtime.h>
#include <hip/hip_bf16.h>

typedef __attribute__((ext_vector_type(2))) float v2f;
typedef __attribute__((ext_vector_type(8))) float v8f;

#define NPTS  2048
#define NTILE 128     // 16-wide tiles per dimension
#define PAIRS 32      // B*S
#define INF_BITS 0x7F800000

__device__ __forceinline__ int imin2(int a, int b) { return a < b ? a : b; }

// One workgroup per (b,s) pair. 256 threads = 8 waves (wave32).
// Per 16x16 tile, one V_WMMA_F32_16X16X4_F32 computes
//   d2 = (-2P | 1) x (T | t2)^T + p2   (K=4: xyz + fused-t2 lane)
// B operand is staged in LDS pre-swizzled per lane-half, so a single
// ds_load_b64 yields the WMMA B register pair directly.
// All min-reductions run on raw f32 bit patterns as signed ints
// (no float canonicalization); final sqrt(max(x,0)) clamps -eps bits.
__global__ __launch_bounds__(256)
void ChamferDistanceLoss_main(const float* __restrict__ pred,
                              const float* __restrict__ targ,
                              float* __restrict__ partial)
{
    __shared__ v2f   bfmt[2 * NPTS]; // [n]=(tx,ty) | [NPTS+n]=(tz,t2)  32 KB
    __shared__ float p2s[NPTS];      // |p|^2                            8 KB
    __shared__ int   minPi[NPTS];    // per-pred min d2 bits             8 KB
    __shared__ int   minTi[NPTS];    // per-target min d2 bits           8 KB
    __shared__ float red[8];

    const int g    = blockIdx.x;
    const int tid  = threadIdx.x;
    const int lane = tid & 31;
    const int wave = tid >> 5;
    const int l16  = lane & 15;
    const int half = lane >> 4;       // 0: lanes 0-15 (K0,K1), 1: lanes 16-31 (K2,K3)

    const float* p = pred + (size_t)g * NPTS * 3;
    const float* t = targ + (size_t)g * NPTS * 3;

    // Stage pre-swizzled B operands (+ norms); pred norms; init column mins.
    for (int i = tid; i < NPTS; i += 256) {
        float tx = t[i*3+0], ty = t[i*3+1], tz = t[i*3+2];
        v2f lo; lo.x = tx; lo.y = ty;
        v2f hi; hi.x = tz; hi.y = tx*tx + ty*ty + tz*tz;
        bfmt[i]        = lo;
        bfmt[NPTS + i] = hi;
        float px = p[i*3+0], py = p[i*3+1], pz = p[i*3+2];
        p2s[i] = px*px + py*py + pz*pz;
        minTi[i] = INF_BITS;
    }
    __syncthreads();

    const int boff = (half ? NPTS : 0) + l16;   // hoisted half-dependent base

    // Each wave owns M-tiles wave, wave+8, ... (16 tiles of 16 pred rows each).
    for (int mt = wave; mt < NTILE; mt += 8) {
        const int mBase = mt * 16;
        const int m = mBase + l16;

        // A (16x4): K0=-2px, K1=-2py, K2=-2pz, K3=1.0 (t2 carrier)
        float px = p[m*3+0], py = p[m*3+1], pz = p[m*3+2];
        v2f a;
        a.x = -2.0f * (half ? pz : px);
        a.y = half ? 1.0f : (-2.0f * py);

        // Loop-invariant C = p2 (8 accumulator registers, reused every WMMA)
        v8f c0;
        #pragma unroll
        for (int k = 0; k < 8; ++k) c0[k] = p2s[mBase + k + 8*half];

        int rmin[8];
        #pragma unroll
        for (int k = 0; k < 8; ++k) rmin[k] = INF_BITS;

        // Software-pipelined B-operand fetch (hide ds_load_b64 latency).
        v2f bm = bfmt[boff];
        #pragma unroll 2
        for (int nt = 0; nt < NTILE; ++nt) {
            v2f bmn = bfmt[boff + (((nt + 1) & (NTILE - 1)) << 4)];  // prefetch

            v8f d2 = __builtin_amdgcn_wmma_f32_16x16x4_f32(
                false, a, false, bm, (short)0, c0, false, false);

            int db[8];
            #pragma unroll
            for (int k = 0; k < 8; ++k) db[k] = __float_as_int(d2[k]);

            // Running row mins (per pred point), integer domain.
            #pragma unroll
            for (int k = 0; k < 8; ++k) rmin[k] = imin2(rmin[k], db[k]);

            // Column partial min over this half's 8 M rows (balanced tree);
            // the LDS atomic merges the two halves and all waves/tiles.
            int cmin = imin2(imin2(imin2(db[0], db[1]), imin2(db[2], db[3])),
                             imin2(imin2(db[4], db[5]), imin2(db[6], db[7])));
            atomicMin(&minTi[(nt << 4) + l16], cmin);

            bm = bmn;
        }

        // Finalize row mins: butterfly within each 16-lane group (cold path).
        #pragma unroll
        for (int step = 1; step <= 8; step <<= 1) {
            #pragma unroll
            for (int k = 0; k < 8; ++k)
                rmin[k] = imin2(rmin[k], __shfl_xor(rmin[k], step, 32));
        }
        if (l16 < 8) {
            int v = rmin[0];
            #pragma unroll
            for (int j = 1; j < 8; ++j) if (l16 == j) v = rmin[j];
            minPi[mBase + l16 + 8*half] = v;   // rows owned exclusively by this wave
        }
    }
    __syncthreads();

    // chamfer[b,s] = (sum_m sqrt(minP) + sum_n sqrt(minT)) / NPTS
    float s = 0.0f;
    for (int i = tid; i < NPTS; i += 256) {
        s += sqrtf(fmaxf(__int_as_float(minPi[i]), 0.0f));
        s += sqrtf(fmaxf(__int_as_float(minTi[i]), 0.0f));
    }
    #pragma unroll
    for (int off = 1; off < 32; off <<= 1)
        s += __shfl_xor(s, off, 32);
    if (lane == 0) red[wave] = s;
    __syncthreads();
    if (tid == 0) {
        float tot = 0.0f;
        #pragma unroll
        for (int w = 0; w < 8; ++w) tot += red[w];
        partial[g] = tot / (float)NPTS;
    }
}

// Deterministic fixed-order final reduction:
// out = mean_b( sum_s chamfer ) / S = (sum over all pairs) / (B*S)
__global__ void ChamferDistanceLoss_finalize(const float* __restrict__ partial,
                                             float* __restrict__ out)
{
    if (threadIdx.x == 0 && blockIdx.x == 0) {
        float tot = 0.0f;
        for (int i = 0; i < PAIRS; ++i) tot += partial[i];
        out[0] = tot / (float)PAIRS;
    }
}

extern "C" void kernel_launch(void* const* d_in, const int* in_sizes, int n_in,
                              void* d_out, int out_size, void* d_ws, size_t ws_size,
                              hipStream_t stream) {
    const float* pred = (const float*)d_in[0];   // [4,8,2048,3] f32
    const float* targ = (const float*)d_in[1];   // [4,8,2048,3] f32
    float* out = (float*)d_out;                  // scalar f32
    float* ws  = (float*)d_ws;                   // >= 32 floats of scratch

    ChamferDistanceLoss_main<<<dim3(PAIRS), dim3(256), 0, stream>>>(pred, targ, ws);
    ChamferDistanceLoss_finalize<<<dim3(1), dim3(32), 0, stream>>>(ws, out);
}